// GraphEmbedder_19490561589476
// MI455X (gfx1250) — compile-verified
//
#include <hip/hip_runtime.h>

typedef __attribute__((ext_vector_type(2))) float v2f;
typedef __attribute__((ext_vector_type(8))) float v8f;

#define N_NODES  50000
#define N_EDGES  800000
#define IN_CH    128
#define HID_CH   256
#define N_GRAPHS 500

// ---------------------------------------------------------------- utilities
__global__ void fill_f32(float* __restrict__ p, float v, int n) {
  int i = blockIdx.x * blockDim.x + threadIdx.x;
  if (i < n) p[i] = v;
}

__global__ void fill_f32x4(float4* __restrict__ p, float v, int n4) {
  int i = blockIdx.x * blockDim.x + threadIdx.x;
  if (i < n4) p[i] = make_float4(v, v, v, v);
}

__global__ void deg_accum(const int* __restrict__ dst, float* __restrict__ deg) {
  int e = blockIdx.x * blockDim.x + threadIdx.x;
  if (e < N_EDGES) atomicAdd(&deg[dst[e]], 1.0f);
}

__global__ void rsqrt_inplace(float* __restrict__ deg) {
  int i = blockIdx.x * blockDim.x + threadIdx.x;
  if (i < N_NODES) deg[i] = rsqrtf(deg[i]);  // deg >= 1 (self-loop)
}

// ---------------------------------------------------------------- W pre-pack
// Rewrite W[K,256] (row major) into WMMA B-fragment order so the GEMM's inner
// loop reads each lane's (K, K+1) pair as one contiguous b64.
// Fragment f = t*16 + c  (t = K-step k0/4, c = 16-col tile), 32 lanes, 2 f32:
//   lane = h*16 + l  ->  { W[(4t+2h)*256 + 16c + l], W[(4t+2h+1)*256 + 16c + l] }
__global__ void pack_w(const float* __restrict__ W, v2f* __restrict__ Bp, int K) {
  int idx = blockIdx.x * blockDim.x + threadIdx.x;           // (K/4)*16*32 threads
  if (idx >= (K / 4) * 16 * 32) return;
  const int lane = idx & 31;
  const int c    = (idx >> 5) & 15;
  const int t    = idx >> 9;
  const int h    = lane >> 4;
  const int l    = lane & 15;
  const int ka   = 4 * t + 2 * h;
  v2f b;
  b.x = W[(size_t)ka * HID_CH + c * 16 + l];
  b.y = W[(size_t)(ka + 1) * HID_CH + c * 16 + l];
  Bp[idx] = b;     // idx == (t*16 + c)*32 + lane
}

// ---------------------------------------------------------------- WMMA GEMM
// C[M,256] = op(A)[M,K] @ W[K,256], op(x) = RELUA ? max(x + bias, 0) : x.
// One wave -> 16(M) x 64(N) tile via V_WMMA_F32_16X16X4_F32; A fragment reused
// across 4 WMMAs per K-step; B fragments pre-packed (4 x b64 per step).
// Loop uses pointer-bump induction so unrolled loads fold into immediate
// offsets (A stride 16B, B stride 4096B per K-step).
template <bool RELUA>
__global__ void __launch_bounds__(256)
gemm_wmma_f32(const float* __restrict__ A, const v2f* __restrict__ Bp,
              const float* __restrict__ bias, float* __restrict__ C,
              int M, int K) {
  const int NN   = HID_CH;            // 256
  const int wave = threadIdx.x >> 5;  // 0..7
  const int lane = threadIdx.x & 31;
  const int half = lane >> 4;
  const int l    = lane & 15;
  const int mTile = blockIdx.x * 2 + (wave >> 2);  // 16-row tile index
  const int nGrp  = wave & 3;                      // 64-col group
  if (mTile * 16 >= M) return;                     // wave-uniform: EXEC stays all-1s

  const int row = mTile * 16 + l;
  const float* __restrict__ aPtr = A + (size_t)row * K + 2 * half;
  const float* __restrict__ bsPtr = RELUA ? (bias + 2 * half) : nullptr;
  const v2f*   __restrict__ bPtr = Bp + (size_t)(nGrp * 4) * 32 + lane;

  v8f acc0 = {}, acc1 = {}, acc2 = {}, acc3 = {};
  for (int t = 0; t < K / 4; ++t) {
    float a0 = aPtr[0], a1 = aPtr[1];
    if (RELUA) {  // fused relu(h + b1) feeding layer-2 GEMM
      a0 = fmaxf(a0 + bsPtr[0], 0.0f);
      a1 = fmaxf(a1 + bsPtr[1], 0.0f);
      bsPtr += 4;
    }
    v2f a; a.x = a0; a.y = a1;

    v2f b0 = bPtr[0], b1 = bPtr[32], b2 = bPtr[64], b3 = bPtr[96];
    aPtr += 4;
    bPtr += 512;   // 16 fragments * 32 lanes per K-step

    acc0 = __builtin_amdgcn_wmma_f32_16x16x4_f32(false, a, false, b0, (short)0, acc0, false, false);
    acc1 = __builtin_amdgcn_wmma_f32_16x16x4_f32(false, a, false, b1, (short)0, acc1, false, false);
    acc2 = __builtin_amdgcn_wmma_f32_16x16x4_f32(false, a, false, b2, (short)0, acc2, false, false);
    acc3 = __builtin_amdgcn_wmma_f32_16x16x4_f32(false, a, false, b3, (short)0, acc3, false, false);
  }

  // C/D layout: VGPR r -> row mTile*16 + r + 8*half, col nGrp*64 + {0,16,32,48} + l
  const int crow  = mTile * 16 + 8 * half;
  const int nBase = nGrp * 64;
#pragma unroll
  for (int r = 0; r < 8; ++r) {
    float* __restrict__ Crow = C + (size_t)(crow + r) * NN + nBase + l;
    Crow[0]  = acc0[r];
    Crow[16] = acc1[r];
    Crow[32] = acc2[r];
    Crow[48] = acc3[r];
  }
}

// ---------------------------------------------------------------- scatter
// One wave per (edge or self-loop): Hout[dst] += dis[src]*dis[dst] * Hin[src]
// 32 lanes x 8 channels; gathers/atomics land in the 192MB L2 (51MB buffers).
__global__ void __launch_bounds__(256)
scatter_edges(const float* __restrict__ Hin, float* __restrict__ Hout,
              const int* __restrict__ src, const int* __restrict__ dst,
              const float* __restrict__ dis) {
  const int e    = blockIdx.x * 8 + (threadIdx.x >> 5);
  const int lane = threadIdx.x & 31;
  if (e >= N_EDGES + N_NODES) return;

  int s, d; float w;
  if (e < N_EDGES) {
    s = src[e]; d = dst[e];
    w = dis[s] * dis[d];
  } else {                       // self loop
    s = e - N_EDGES; d = s;
    float r = dis[s]; w = r * r;
  }

  const float4* __restrict__ in = (const float4*)(Hin + (size_t)s * HID_CH);
  float4 v0 = in[lane];          // channels [4*lane .. 4*lane+3]
  float4 v1 = in[lane + 32];     // channels [128+4*lane ..]
  float* __restrict__ out = Hout + (size_t)d * HID_CH;
  const int c0 = lane * 4;
  atomicAdd(out + c0 + 0,   v0.x * w);
  atomicAdd(out + c0 + 1,   v0.y * w);
  atomicAdd(out + c0 + 2,   v0.z * w);
  atomicAdd(out + c0 + 3,   v0.w * w);
  atomicAdd(out + c0 + 128, v1.x * w);
  atomicAdd(out + c0 + 129, v1.y * w);
  atomicAdd(out + c0 + 130, v1.z * w);
  atomicAdd(out + c0 + 131, v1.w * w);
}

// ---------------------------------------------------------------- pooling
// out[batch[i]] += relu(H[i] + b2); cnt[batch[i]] += 1  (one wave per node)
__global__ void __launch_bounds__(256)
pool_nodes(const float* __restrict__ H, const float* __restrict__ bias,
           const int* __restrict__ batch, float* __restrict__ out,
           float* __restrict__ cnt) {
  const int i    = blockIdx.x * 8 + (threadIdx.x >> 5);
  const int lane = threadIdx.x & 31;
  if (i >= N_NODES) return;
  const int g = batch[i];
  const float* __restrict__ h = H + (size_t)i * HID_CH;
  float* __restrict__ o = out + (size_t)g * HID_CH;
#pragma unroll
  for (int j = 0; j < 8; ++j) {
    const int c = j * 32 + lane;            // coalesced
    atomicAdd(o + c, fmaxf(h[c] + bias[c], 0.0f));
  }
  if (lane == 0) atomicAdd(cnt + g, 1.0f);
}

__global__ void finalize_mean(float* __restrict__ out, const float* __restrict__ cnt) {
  int idx = blockIdx.x * blockDim.x + threadIdx.x;
  if (idx < N_GRAPHS * HID_CH) out[idx] /= fmaxf(cnt[idx / HID_CH], 1.0f);
}

// ---------------------------------------------------------------- launcher
extern "C" void kernel_launch(void* const* d_in, const int* in_sizes, int n_in,
                              void* d_out, int out_size, void* d_ws, size_t ws_size,
                              hipStream_t stream) {
  const float* x   = (const float*)d_in[0];   // [50000,128]
  const float* W1  = (const float*)d_in[1];   // [128,256]
  const float* b1  = (const float*)d_in[2];   // [256]
  const float* W2  = (const float*)d_in[3];   // [256,256]
  const float* b2  = (const float*)d_in[4];   // [256]
  const int*   ei  = (const int*)d_in[5];     // [2,800000]
  const int*   bat = (const int*)d_in[6];     // [50000]
  const int* src = ei;
  const int* dst = ei + N_EDGES;
  float* out = (float*)d_out;                 // [500,256]

  // workspace layout (256B aligned)
  char* ws = (char*)d_ws;
  float* dis  = (float*)(ws);                                   // 50000 f32
  float* bufA = (float*)(ws + 200192);                          // 50000*256 f32
  float* bufB = (float*)(ws + 200192 + 51200000ull);            // 50000*256 f32
  float* cnt  = (float*)(ws + 200192 + 2 * 51200000ull);        // 500 f32 (pad 2048)
  v2f* pW1    = (v2f*)(ws + 200192 + 2 * 51200000ull + 2048);   // 128*256 f32
  v2f* pW2    = (v2f*)(ws + 200192 + 2 * 51200000ull + 2048 + 131072);  // 256*256 f32
  const int NH = N_NODES * HID_CH;                              // 12.8M

  const int T = 256;
  // degree (with self-loop) -> D^{-1/2}; pack weights into fragment order
  fill_f32<<<(N_NODES + T - 1) / T, T, 0, stream>>>(dis, 1.0f, N_NODES);
  deg_accum<<<(N_EDGES + T - 1) / T, T, 0, stream>>>(dst, dis);
  rsqrt_inplace<<<(N_NODES + T - 1) / T, T, 0, stream>>>(dis);
  pack_w<<<(IN_CH * 128 + T - 1) / T, T, 0, stream>>>(W1, pW1, IN_CH);
  pack_w<<<(HID_CH * 128 + T - 1) / T, T, 0, stream>>>(W2, pW2, HID_CH);

  const int mTiles = N_NODES / 16;                   // 3125
  const int gemmBlocks = (mTiles + 1) / 2;           // 2 M-tiles per 8-wave block
  const int edgeBlocks = (N_EDGES + N_NODES + 7) / 8;

  // layer 1: bufA = x @ W1 ; bufB = scatter(norm * bufA[src])
  gemm_wmma_f32<false><<<gemmBlocks, T, 0, stream>>>(x, pW1, nullptr, bufA, N_NODES, IN_CH);
  fill_f32x4<<<(NH / 4 + T - 1) / T, T, 0, stream>>>((float4*)bufB, 0.0f, NH / 4);
  scatter_edges<<<edgeBlocks, T, 0, stream>>>(bufA, bufB, src, dst, dis);

  // layer 2: bufA = relu(bufB + b1) @ W2 ; bufB = scatter(norm * bufA[src])
  gemm_wmma_f32<true><<<gemmBlocks, T, 0, stream>>>(bufB, pW2, b1, bufA, N_NODES, HID_CH);
  fill_f32x4<<<(NH / 4 + T - 1) / T, T, 0, stream>>>((float4*)bufB, 0.0f, NH / 4);
  scatter_edges<<<edgeBlocks, T, 0, stream>>>(bufA, bufB, src, dst, dis);

  // mean pool of relu(bufB + b2) over batch ids
  fill_f32<<<(N_GRAPHS * HID_CH + T - 1) / T, T, 0, stream>>>(out, 0.0f, N_GRAPHS * HID_CH);
  fill_f32<<<(N_GRAPHS + T - 1) / T, T, 0, stream>>>(cnt, 0.0f, N_GRAPHS);
  pool_nodes<<<(N_NODES + 7) / 8, T, 0, stream>>>(bufB, b2, bat, out, cnt);
  finalize_mean<<<(N_GRAPHS * HID_CH + T - 1) / T, T, 0, stream>>>(out, cnt);
}